// GraphConvNet_87136296501509
// MI455X (gfx1250) — compile-verified
//
#include <hip/hip_runtime.h>
#include <hip/hip_bf16.h>

#define N_NODES 100000
#define N_EDGES 1600000
#define N_GRAPH 100
#define IN_DIM  64
#define LATENT  128
#define LN_EPS  1e-5f

typedef __attribute__((ext_vector_type(2))) float v2f;
typedef __attribute__((ext_vector_type(8))) float v8f;

// ---------------------------------------------------------------------------
// Generic tiled GEMM:  Out[n x 128] = act(A[n x K] @ W[K x 128] + bias) * scale
// One wave -> 16x32 output tile via two V_WMMA_F32_16X16X4_F32 accumulators.
// Block = 256 threads (8 waves) -> 32 rows x 128 cols. grid.x = n/32.
// n must be a multiple of 32 (100000 is), so EXEC is all-ones at every WMMA.
// ---------------------------------------------------------------------------
__global__ __launch_bounds__(256)
void gcn_gemm_kernel(const float* __restrict__ A,
                     const float* __restrict__ W,
                     const float* __restrict__ bias,
                     const float* __restrict__ row_scale,  // may be nullptr
                     float* __restrict__ Out,
                     int K, int relu)
{
    const int lane    = threadIdx.x & 31;
    const int wave    = threadIdx.x >> 5;
    const int r0      = (blockIdx.x * 2 + (wave >> 2)) * 16;   // 16-row tile base
    const int colBase = (wave & 3) * 32;                        // 32-col group
    const int m       = r0 + (lane & 15);                       // A row for this lane
    const int kh      = (lane >> 4) << 1;                       // K sub-pair: 0 or 2
    const int col     = colBase + (lane & 15);                  // B/D column

    v8f acc0 = {};
    v8f acc1 = {};

    const float* arow = A + (size_t)m * K + kh;
    for (int k = 0; k < K; k += 4) {
        v2f a;
        a.x = arow[k];
        a.y = arow[k + 1];
        const float* wr = W + (size_t)(k + kh) * LATENT;
        v2f b0, b1;
        b0.x = wr[col];
        b0.y = wr[LATENT + col];
        b1.x = wr[col + 16];
        b1.y = wr[LATENT + col + 16];
        acc0 = __builtin_amdgcn_wmma_f32_16x16x4_f32(false, a, false, b0,
                                                     (short)0, acc0, false, false);
        acc1 = __builtin_amdgcn_wmma_f32_16x16x4_f32(false, a, false, b1,
                                                     (short)0, acc1, false, false);
    }

    // Epilogue: bias + optional relu + optional per-row scale, then store.
    const int   crow = r0 + ((lane >> 4) << 3);
    const float bv0  = bias[col];
    const float bv1  = bias[col + 16];
#pragma unroll
    for (int i = 0; i < 8; ++i) {
        const int rr = crow + i;
        float v0 = acc0[i] + bv0;
        float v1 = acc1[i] + bv1;
        if (relu) { v0 = fmaxf(v0, 0.0f); v1 = fmaxf(v1, 0.0f); }
        if (row_scale) { const float s = row_scale[rr]; v0 *= s; v1 *= s; }
        Out[(size_t)rr * LATENT + col]      = v0;
        Out[(size_t)rr * LATENT + col + 16] = v1;
    }
}

// ---------------------------------------------------------------------------
// Degree / utility kernels
// ---------------------------------------------------------------------------
__global__ void gcn_fill_kernel(float* __restrict__ p, float v, int n)
{
    int i = blockIdx.x * blockDim.x + threadIdx.x;
    if (i < n) p[i] = v;
}

__global__ void gcn_edge_deg_kernel(const int* __restrict__ snd,
                                    const int* __restrict__ rcv,
                                    float* __restrict__ sdeg,
                                    float* __restrict__ rdeg, int ne)
{
    int e = blockIdx.x * blockDim.x + threadIdx.x;
    if (e < ne) {
        atomicAdd(sdeg + snd[e], 1.0f);
        atomicAdd(rdeg + rcv[e], 1.0f);
    }
}

__global__ void gcn_rsqrt_kernel(float* __restrict__ p, int n)
{
    int i = blockIdx.x * blockDim.x + threadIdx.x;
    if (i < n) p[i] = rsqrtf(fmaxf(p[i], 1.0f));
}

// ---------------------------------------------------------------------------
// Edge scatter: one wave per edge, 4 feats per lane (float4 gather + atomics)
// ---------------------------------------------------------------------------
__global__ __launch_bounds__(256)
void gcn_scatter_kernel(const float* __restrict__ x,
                        const int* __restrict__ snd,
                        const int* __restrict__ rcv,
                        float* __restrict__ agg, int ne)
{
    int gid  = blockIdx.x * blockDim.x + threadIdx.x;
    int e    = gid >> 5;
    int lane = gid & 31;
    if (e >= ne) return;
    const int s = snd[e];
    const int r = rcv[e];
    float4 v = ((const float4*)(x + (size_t)s * LATENT))[lane];
    float* dst = agg + (size_t)r * LATENT + lane * 4;
    atomicAdd(dst + 0, v.x);
    atomicAdd(dst + 1, v.y);
    atomicAdd(dst + 2, v.z);
    atomicAdd(dst + 3, v.w);
}

// ---------------------------------------------------------------------------
// h = LayerNorm((agg + x_self)*inv_rdeg + h); one wave per node
// ---------------------------------------------------------------------------
__global__ __launch_bounds__(256)
void gcn_combine_ln_kernel(const float* __restrict__ agg,
                           const float* __restrict__ x,
                           const float* __restrict__ ird,
                           float* __restrict__ h,
                           const float* __restrict__ lns,
                           const float* __restrict__ lno, int n)
{
    int node = blockIdx.x * 8 + (threadIdx.x >> 5);
    int lane = threadIdx.x & 31;
    if (node >= n) return;
    const size_t base = (size_t)node * LATENT + lane * 4;
    const float4 va = *(const float4*)(agg + base);
    const float4 vx = *(const float4*)(x + base);
    const float4 vh = *(const float4*)(h + base);
    const float  s  = ird[node];
    float t0 = (va.x + vx.x) * s + vh.x;
    float t1 = (va.y + vx.y) * s + vh.y;
    float t2 = (va.z + vx.z) * s + vh.z;
    float t3 = (va.w + vx.w) * s + vh.w;

    float sum = t0 + t1 + t2 + t3;
    float sq  = t0 * t0 + t1 * t1 + t2 * t2 + t3 * t3;
#pragma unroll
    for (int off = 16; off > 0; off >>= 1) {
        sum += __shfl_xor(sum, off, 32);
        sq  += __shfl_xor(sq,  off, 32);
    }
    const float mu   = sum * (1.0f / LATENT);
    const float var  = sq * (1.0f / LATENT) - mu * mu;
    const float rstd = rsqrtf(var + LN_EPS);

    const int f = lane * 4;
    float4 o;
    o.x = (t0 - mu) * rstd * lns[f + 0] + lno[f + 0];
    o.y = (t1 - mu) * rstd * lns[f + 1] + lno[f + 1];
    o.z = (t2 - mu) * rstd * lns[f + 2] + lno[f + 2];
    o.w = (t3 - mu) * rstd * lns[f + 3] + lno[f + 3];
    *(float4*)(h + base) = o;
}

// ---------------------------------------------------------------------------
// Segment-sum pooling: one wave per node
// ---------------------------------------------------------------------------
__global__ __launch_bounds__(256)
void gcn_pool_kernel(const float* __restrict__ h,
                     const int* __restrict__ gids,
                     float* __restrict__ pooled,
                     float* __restrict__ cnt, int n)
{
    int node = blockIdx.x * 8 + (threadIdx.x >> 5);
    int lane = threadIdx.x & 31;
    if (node >= n) return;
    const int g = gids[node];
    const float4 v = *(const float4*)(h + (size_t)node * LATENT + lane * 4);
    float* d = pooled + (size_t)g * LATENT + lane * 4;
    atomicAdd(d + 0, v.x);
    atomicAdd(d + 1, v.y);
    atomicAdd(d + 2, v.z);
    atomicAdd(d + 3, v.w);
    if (lane == 0) atomicAdd(cnt + g, 1.0f);
}

// ---------------------------------------------------------------------------
// Decoder: out[g] = (pooled[g]/cnt[g]) . dec_w + dec_b ; one wave per graph
// ---------------------------------------------------------------------------
__global__ __launch_bounds__(32)
void gcn_decode_kernel(const float* __restrict__ pooled,
                       const float* __restrict__ cnt,
                       const float* __restrict__ dw,
                       const float* __restrict__ db,
                       float* __restrict__ out)
{
    const int g    = blockIdx.x;
    const int lane = threadIdx.x;
    const float inv = 1.0f / fmaxf(cnt[g], 1.0f);
    float s = 0.0f;
#pragma unroll
    for (int j = 0; j < 4; ++j) {
        const int f = lane * 4 + j;
        s += pooled[(size_t)g * LATENT + f] * dw[f];
    }
    s *= inv;
#pragma unroll
    for (int off = 16; off > 0; off >>= 1) s += __shfl_xor(s, off, 32);
    if (lane == 0) out[g] = s + db[0];
}

// ---------------------------------------------------------------------------
extern "C" void kernel_launch(void* const* d_in, const int* in_sizes, int n_in,
                              void* d_out, int out_size, void* d_ws, size_t ws_size,
                              hipStream_t stream)
{
    const float* nodes   = (const float*)d_in[0];
    const int*   snd     = (const int*)  d_in[1];
    const int*   rcv     = (const int*)  d_in[2];
    const int*   gids    = (const int*)  d_in[3];
    const float* embed_w = (const float*)d_in[4];
    const float* embed_b = (const float*)d_in[5];
    const float* mlp_w   = (const float*)d_in[6];  // [2,2,128,128]
    const float* mlp_b   = (const float*)d_in[7];  // [2,2,128]
    const float* lns     = (const float*)d_in[8];  // [2,128]
    const float* lno     = (const float*)d_in[9];  // [2,128]
    const float* dw      = (const float*)d_in[10]; // [128,1]
    const float* db      = (const float*)d_in[11]; // [1]
    float*       out     = (float*)d_out;          // [100,1]

    float* ws  = (float*)d_ws;
    float* h      = ws;                                  // N_NODES*128
    float* x1     = h  + (size_t)N_NODES * LATENT;       // N_NODES*128 (also agg)
    float* x2     = x1 + (size_t)N_NODES * LATENT;       // N_NODES*128
    float* isd    = x2 + (size_t)N_NODES * LATENT;       // N_NODES
    float* ird    = isd + N_NODES;                       // N_NODES
    float* pooled = ird + N_NODES;                       // N_GRAPH*128
    float* cnt    = pooled + (size_t)N_GRAPH * LATENT;   // N_GRAPH

    const int T = 256;
    // Degrees (init to 1.0 for the self-loop, then count edges, then rsqrt).
    gcn_fill_kernel<<<(N_NODES + T - 1) / T, T, 0, stream>>>(isd, 1.0f, N_NODES);
    gcn_fill_kernel<<<(N_NODES + T - 1) / T, T, 0, stream>>>(ird, 1.0f, N_NODES);
    gcn_edge_deg_kernel<<<(N_EDGES + T - 1) / T, T, 0, stream>>>(snd, rcv, isd, ird, N_EDGES);
    gcn_rsqrt_kernel<<<(N_NODES + T - 1) / T, T, 0, stream>>>(isd, N_NODES);
    gcn_rsqrt_kernel<<<(N_NODES + T - 1) / T, T, 0, stream>>>(ird, N_NODES);

    // Embed: h = nodes @ embed_w + embed_b   (WMMA, K=64)
    gcn_gemm_kernel<<<N_NODES / 32, T, 0, stream>>>(nodes, embed_w, embed_b,
                                                    nullptr, h, IN_DIM, 0);

    for (int s = 0; s < 2; ++s) {
        const float* w0 = mlp_w + (size_t)(s * 2 + 0) * LATENT * LATENT;
        const float* w1 = mlp_w + (size_t)(s * 2 + 1) * LATENT * LATENT;
        const float* b0 = mlp_b + (size_t)(s * 2 + 0) * LATENT;
        const float* b1 = mlp_b + (size_t)(s * 2 + 1) * LATENT;

        // MLP layer 0: x1 = relu(h @ w0 + b0)
        gcn_gemm_kernel<<<N_NODES / 32, T, 0, stream>>>(h, w0, b0, nullptr, x1, LATENT, 1);
        // MLP layer 1 (+ sender-degree scale fused): x2 = relu(x1 @ w1 + b1) * isd
        gcn_gemm_kernel<<<N_NODES / 32, T, 0, stream>>>(x1, w1, b1, isd, x2, LATENT, 1);

        // Edge aggregation into x1 (reused as agg buffer).
        gcn_fill_kernel<<<((size_t)N_NODES * LATENT + T - 1) / T, T, 0, stream>>>(
            x1, 0.0f, N_NODES * LATENT);
        gcn_scatter_kernel<<<((size_t)N_EDGES * 32 + T - 1) / T, T, 0, stream>>>(
            x2, snd, rcv, x1, N_EDGES);

        // h = LN((agg + x_self)*ird + h)  (self-loop edge fused here)
        gcn_combine_ln_kernel<<<N_NODES / 8, T, 0, stream>>>(
            x1, x2, ird, h, lns + s * LATENT, lno + s * LATENT, N_NODES);
    }

    // Pool + decode.
    gcn_fill_kernel<<<(N_GRAPH * LATENT + T - 1) / T, T, 0, stream>>>(pooled, 0.0f, N_GRAPH * LATENT);
    gcn_fill_kernel<<<1, T, 0, stream>>>(cnt, 0.0f, N_GRAPH);
    gcn_pool_kernel<<<N_NODES / 8, T, 0, stream>>>(h, gids, pooled, cnt, N_NODES);
    gcn_decode_kernel<<<N_GRAPH, 32, 0, stream>>>(pooled, cnt, dw, db, out);
}